// Attn_39513699123349
// MI455X (gfx1250) — compile-verified
//
#include <hip/hip_runtime.h>

#define S_LEN 4096
#define BATCH 8
#define HDIM  1024

typedef __attribute__((ext_vector_type(2))) float v2f;
typedef __attribute__((ext_vector_type(8))) float v8f;

// ---------------------------------------------------------------------------
// Kernel 1: q[b,h] = sum_o hidden[b,o] * W[o,h]   via V_WMMA_F32_16X16X4_F32
// D tile = 16 h-rows x 16 b-cols (b cols 8..15 zero-padded).
//   A[m=h, k=o] = W[o*H + h]        (16x4 f32 fragment)
//   B[k=o, n=b] = hidden[b*H + o]   (4x16 f32 fragment, n>=8 -> 0)
// fp32 A-frag layout (ISA 7.12.2): M = lane&15; lane-half selects K pair
// {0,1} vs {2,3}; VGPR0=K_even, VGPR1=K_odd. B/D: N striped across lanes.
// ---------------------------------------------------------------------------
__global__ void __launch_bounds__(32)
proj_q_wmma(const float* __restrict__ W, const float* __restrict__ hidden,
            float* __restrict__ q) {
  const int h0   = blockIdx.x * 16;
  const int lane = threadIdx.x & 31;
  const int half = lane >> 4;   // selects K offset pair
  const int l    = lane & 15;   // M for A-frag, N for B/D frags
  const int bl   = l & 7;       // clamped batch index for B-frag loads

  v8f acc = {};
  for (int k = 0; k < HDIM; k += 4) {
    const int kb = k + half * 2;
    v2f a, bf;
    a.x = W[(size_t)(kb + 0) * HDIM + h0 + l];
    a.y = W[(size_t)(kb + 1) * HDIM + h0 + l];
    const float b0 = hidden[bl * HDIM + kb + 0];
    const float b1 = hidden[bl * HDIM + kb + 1];
    bf.x = (l < 8) ? b0 : 0.0f;
    bf.y = (l < 8) ? b1 : 0.0f;
    acc = __builtin_amdgcn_wmma_f32_16x16x4_f32(
        /*neg_a=*/false, a, /*neg_b=*/false, bf,
        /*c_mod=*/(short)0, acc, /*reuse_a=*/false, /*reuse_b=*/false);
  }

  // D layout: VGPR v -> M = v + 8*half, N = lane&15
  if (l < BATCH) {
#pragma unroll
    for (int v = 0; v < 8; ++v) {
      const int m = v + half * 8;
      q[(size_t)l * HDIM + h0 + m] = acc[v];
    }
  }
}

// ---------------------------------------------------------------------------
// Kernel 2 (bandwidth-bound, 134 MB stream): e[b,s] = q[b] . enc[s,b,:]
// One wave per (s,b) row; coalesced float4 loads; cross-lane shuffle reduce.
// ---------------------------------------------------------------------------
__global__ void __launch_bounds__(256)
dot_rows(const float* __restrict__ enc, const float* __restrict__ q,
         float* __restrict__ e) {
  const int wave = threadIdx.x >> 5;
  const int lane = threadIdx.x & 31;
  const int row  = blockIdx.x * 8 + wave;   // row = s*BATCH + b
  const int b    = row & (BATCH - 1);
  const int s    = row >> 3;

  const float4* ep = (const float4*)(enc + (size_t)row * HDIM);
  const float4* qp = (const float4*)(q + (size_t)b * HDIM);

  float acc = 0.0f;
#pragma unroll
  for (int i = 0; i < HDIM / 4 / 32; ++i) {   // 8 iterations
    const float4 ev = ep[i * 32 + lane];
    const float4 qv = qp[i * 32 + lane];
    acc = fmaf(ev.x, qv.x, acc);
    acc = fmaf(ev.y, qv.y, acc);
    acc = fmaf(ev.z, qv.z, acc);
    acc = fmaf(ev.w, qv.w, acc);
  }
#pragma unroll
  for (int off = 16; off > 0; off >>= 1) acc += __shfl_xor(acc, off, 32);
  if (lane == 0) e[(size_t)b * S_LEN + s] = acc;
}

// ---------------------------------------------------------------------------
// Kernel 3: softmax over s for each batch row (8 rows x 4096).
// ---------------------------------------------------------------------------
__global__ void __launch_bounds__(256)
softmax_rows(const float* __restrict__ e, float* __restrict__ out) {
  const int b = blockIdx.x;
  const float* row  = e + (size_t)b * S_LEN;
  float*       orow = out + (size_t)b * S_LEN;

  __shared__ float lds[8];
  __shared__ float s_max, s_sum;
  const int tid = threadIdx.x;
  const int lane = tid & 31, wave = tid >> 5;

  float m = -3.402823466e38f;
  for (int i = tid; i < S_LEN; i += 256) m = fmaxf(m, row[i]);
#pragma unroll
  for (int off = 16; off > 0; off >>= 1) m = fmaxf(m, __shfl_xor(m, off, 32));
  if (lane == 0) lds[wave] = m;
  __syncthreads();
  if (wave == 0) {
    float t = (lane < 8) ? lds[lane] : -3.402823466e38f;
#pragma unroll
    for (int off = 4; off > 0; off >>= 1) t = fmaxf(t, __shfl_xor(t, off, 32));
    if (lane == 0) s_max = t;
  }
  __syncthreads();
  const float mx = s_max;

  float sum = 0.0f;
  for (int i = tid; i < S_LEN; i += 256) sum += __expf(row[i] - mx);
#pragma unroll
  for (int off = 16; off > 0; off >>= 1) sum += __shfl_xor(sum, off, 32);
  if (lane == 0) lds[wave] = sum;
  __syncthreads();
  if (wave == 0) {
    float t = (lane < 8) ? lds[lane] : 0.0f;
#pragma unroll
    for (int off = 4; off > 0; off >>= 1) t += __shfl_xor(t, off, 32);
    if (lane == 0) s_sum = t;
  }
  __syncthreads();
  const float inv = 1.0f / s_sum;
  for (int i = tid; i < S_LEN; i += 256) orow[i] = __expf(row[i] - mx) * inv;
}

// ---------------------------------------------------------------------------
extern "C" void kernel_launch(void* const* d_in, const int* in_sizes, int n_in,
                              void* d_out, int out_size, void* d_ws, size_t ws_size,
                              hipStream_t stream) {
  const float* hidden = (const float*)d_in[0];  // [1,B,H]
  const float* enc    = (const float*)d_in[1];  // [S,B,H]
  const float* attn_w = (const float*)d_in[2];  // [H,H]
  // d_in[3] = attn_b: constant shift per batch row -> cancels in softmax.

  float* out = (float*)d_out;                   // [B,1,S]
  float* q   = (float*)d_ws;                    // B*H   floats (32 KB)
  float* e   = q + (size_t)BATCH * HDIM;        // B*S   floats (128 KB)

  proj_q_wmma<<<HDIM / 16, 32, 0, stream>>>(attn_w, hidden, q);
  dot_rows<<<(S_LEN * BATCH) / 8, 256, 0, stream>>>(enc, q, e);
  softmax_rows<<<BATCH, 256, 0, stream>>>(e, out);
}